// PathwayGraphEmbedding_61856118997221
// MI455X (gfx1250) — compile-verified
//
#include <hip/hip_runtime.h>

typedef __attribute__((ext_vector_type(2))) float v2f;
typedef __attribute__((ext_vector_type(8))) float v8f;

#define Bq 32
#define Nn 20000
#define Ee 320000
#define Dd 128
#define Mrows ((size_t)Bq * Nn)   // 640000 rows of the flattened [B*N, 128] matrix

// ---------------- graph normalization ----------------
__global__ void k_init_deg(float* __restrict__ deg) {
    int n = blockIdx.x * blockDim.x + threadIdx.x;
    if (n < Nn) deg[n] = 1.0f;                     // self-loop
}

__global__ void k_count_deg(const int* __restrict__ dst, float* __restrict__ deg) {
    int e = blockIdx.x * blockDim.x + threadIdx.x;
    if (e < Ee) atomicAdd(&deg[dst[e]], 1.0f);
}

__global__ void k_norms(const float* __restrict__ deg,
                        float* __restrict__ dinv, float* __restrict__ self_n) {
    int n = blockIdx.x * blockDim.x + threadIdx.x;
    if (n < Nn) {
        float d = deg[n];
        dinv[n]   = rsqrtf(d);
        self_n[n] = 1.0f / d;
    }
}

__global__ void k_edge_norm(const int* __restrict__ src, const int* __restrict__ dst,
                            const float* __restrict__ dinv, float* __restrict__ enorm) {
    int e = blockIdx.x * blockDim.x + threadIdx.x;
    if (e < Ee) enorm[e] = dinv[src[e]] * dinv[dst[e]];
}

// ---------------- GEMM: H[M,128] = X[M,128] @ W[128,128] via f32 WMMA ----------------
// One wave -> one 16x16 output tile. Block = 8 waves = 16 rows x 128 cols.
__global__ __launch_bounds__(256) void k_gemm_wmma(const float* __restrict__ X,
                                                   const float* __restrict__ W,
                                                   float* __restrict__ H) {
    const int wave  = threadIdx.x >> 5;          // 0..7 -> 16-col tile
    const int lane  = threadIdx.x & 31;
    const int l16   = lane & 15;
    const int hi    = lane >> 4;                 // 0 or 1
    const size_t r0 = (size_t)blockIdx.x * 16;   // row tile base
    const int c0    = wave * 16;                 // col tile base

    const float* __restrict__ xrow = X + (r0 + (size_t)l16) * Dd; // A: m = lane&15
    v8f acc = {};

#pragma unroll
    for (int k0 = 0; k0 < Dd; k0 += 4) {
        const int ka = k0 + 2 * hi;
        // A 16x4 fragment: v0 = A[m][ka], v1 = A[m][ka+1]  (contiguous -> float2 load)
        v2f a = *(const v2f*)(xrow + ka);
        // B 4x16 fragment: v0 = W[ka][c], v1 = W[ka+1][c], c = c0 + lane&15
        v2f b;
        b.x = W[(size_t)ka       * Dd + c0 + l16];
        b.y = W[(size_t)(ka + 1) * Dd + c0 + l16];
        acc = __builtin_amdgcn_wmma_f32_16x16x4_f32(
            /*neg_a=*/false, a, /*neg_b=*/false, b,
            /*c_mod=*/(short)0, acc, /*reuse_a=*/false, /*reuse_b=*/false);
    }

    // C/D layout: VGPR j -> row j (lanes 0-15) / row j+8 (lanes 16-31), col = lane&15
#pragma unroll
    for (int j = 0; j < 8; ++j) {
        size_t row = r0 + (size_t)(j + 8 * hi);
        H[row * Dd + c0 + l16] = acc[j];
    }
}

// ---------------- layer-1 combine init: X = self_norm*H + bias ----------------
__global__ void k_initC(const float* __restrict__ H, const float* __restrict__ self_n,
                        const float* __restrict__ bias, float* __restrict__ X) {
    size_t i = (size_t)blockIdx.x * blockDim.x + threadIdx.x;   // over B*N*D
    int f = (int)(i & (Dd - 1));
    int n = (int)((i >> 7) % Nn);
    X[i] = self_n[n] * H[i] + bias[f];
}

// ---------------- layer-1 edge scatter: X[b,dst,:] += enorm * H[b,src,:] ----------------
__global__ __launch_bounds__(128) void k_scatter(const float* __restrict__ H,
                                                 const int* __restrict__ src,
                                                 const int* __restrict__ dst,
                                                 const float* __restrict__ enorm,
                                                 float* __restrict__ X) {
    const int f = threadIdx.x;                   // feature
    const size_t base = (size_t)blockIdx.y * Nn * Dd;
    const int e0 = blockIdx.x * 32;
#pragma unroll 4
    for (int i = 0; i < 32; ++i) {
        int e = e0 + i;
        int s = src[e], d = dst[e];
        float w = enorm[e];
        float v = w * H[base + ((size_t)s << 7) + f];
        atomicAdd(&X[base + ((size_t)d << 7) + f], v);
    }
}

__global__ void k_relu(float* __restrict__ X) {
    size_t i = (size_t)blockIdx.x * blockDim.x + threadIdx.x;
    X[i] = fmaxf(X[i], 0.0f);
}

// ---------------- layer-2 reductions into acc[B,128] ----------------
__global__ void k_zero_acc(float* __restrict__ acc) {
    int i = blockIdx.x * blockDim.x + threadIdx.x;
    if (i < Bq * Dd) acc[i] = 0.0f;
}

__global__ __launch_bounds__(128) void k_reduce_self(const float* __restrict__ H,
                                                     const float* __restrict__ self_n,
                                                     float* __restrict__ acc) {
    const int f = threadIdx.x;
    const int b = blockIdx.y;
    const size_t base = (size_t)b * Nn * Dd;
    const int n0 = blockIdx.x * 200;
    float s = 0.0f;
    for (int i = 0; i < 200; ++i) {
        int n = n0 + i;
        s += self_n[n] * H[base + ((size_t)n << 7) + f];
    }
    atomicAdd(&acc[b * Dd + f], s);
}

__global__ __launch_bounds__(128) void k_reduce_edges(const float* __restrict__ H,
                                                      const int* __restrict__ src,
                                                      const float* __restrict__ enorm,
                                                      float* __restrict__ acc) {
    const int f = threadIdx.x;
    const int b = blockIdx.y;
    const size_t base = (size_t)b * Nn * Dd;
    const int e0 = blockIdx.x * 500;
    float s = 0.0f;
    for (int i = 0; i < 500; ++i) {
        int e = e0 + i;
        s += enorm[e] * H[base + ((size_t)src[e] << 7) + f];
    }
    atomicAdd(&acc[b * Dd + f], s);
}

__global__ void k_finalize(const float* __restrict__ acc, const float* __restrict__ bias,
                           float* __restrict__ out) {
    int i = blockIdx.x * blockDim.x + threadIdx.x;   // over B*128
    if (i < Bq * Dd) {
        int f = i & (Dd - 1);
        out[i] = acc[i] * (1.0f / (float)Nn) + bias[f];
    }
}

// ---------------- host launch ----------------
extern "C" void kernel_launch(void* const* d_in, const int* in_sizes, int n_in,
                              void* d_out, int out_size, void* d_ws, size_t ws_size,
                              hipStream_t stream) {
    const float* gene = (const float*)d_in[0];          // [B,N,128] == [M,128]
    const int*   ei   = (const int*)d_in[1];            // [2,E]
    const int*   src  = ei;
    const int*   dst  = ei + Ee;
    const float* W1   = (const float*)d_in[2];
    const float* b1   = (const float*)d_in[3];
    const float* W2   = (const float*)d_in[4];
    const float* b2   = (const float*)d_in[5];
    float* out = (float*)d_out;

    float* ws     = (float*)d_ws;
    float* deg    = ws;                  // 20000 (pad 20480)
    float* dinv   = ws + 20480;          // 20000
    float* self_n = ws + 40960;          // 20000
    float* enorm  = ws + 61440;          // 320000
    float* acc    = ws + 381440;         // 4096
    float* Hbuf   = ws + 385536;         // B*N*128
    float* Xbuf   = Hbuf + Mrows * Dd;   // B*N*128

    // graph normalization
    k_init_deg <<<(Nn + 255) / 256, 256, 0, stream>>>(deg);
    k_count_deg<<<(Ee + 255) / 256, 256, 0, stream>>>(dst, deg);
    k_norms    <<<(Nn + 255) / 256, 256, 0, stream>>>(deg, dinv, self_n);
    k_edge_norm<<<(Ee + 255) / 256, 256, 0, stream>>>(src, dst, dinv, enorm);

    const int gemm_blocks = (int)(Mrows / 16);           // 40000
    const int elem_blocks = (int)(Mrows * Dd / 256);     // 320000

    // ---- layer 1 ----
    k_gemm_wmma<<<gemm_blocks, 256, 0, stream>>>(gene, W1, Hbuf);
    k_initC    <<<elem_blocks, 256, 0, stream>>>(Hbuf, self_n, b1, Xbuf);
    dim3 sg(Ee / 32, Bq);
    k_scatter  <<<sg, 128, 0, stream>>>(Hbuf, src, dst, enorm, Xbuf);
    k_relu     <<<elem_blocks, 256, 0, stream>>>(Xbuf);

    // ---- layer 2 (agg never materialized; mean-pool folded into reductions) ----
    k_gemm_wmma<<<gemm_blocks, 256, 0, stream>>>(Xbuf, W2, Hbuf);
    k_zero_acc <<<(Bq * Dd + 255) / 256, 256, 0, stream>>>(acc);
    dim3 rs(Nn / 200, Bq);
    k_reduce_self <<<rs, 128, 0, stream>>>(Hbuf, self_n, acc);
    dim3 re(Ee / 500, Bq);
    k_reduce_edges<<<re, 128, 0, stream>>>(Hbuf, src, enorm, acc);
    k_finalize <<<(Bq * Dd + 255) / 256, 256, 0, stream>>>(acc, b2, out);
}